// Sinkhorn_62466004353554
// MI455X (gfx1250) — compile-verified
//
#include <hip/hip_runtime.h>
#include <cstdint>
#include <cstddef>

// Sinkhorn with row/col potentials: u[i], v[j] per batch.
// N = M = 1024, iterations = 10 x (row pass, col pass), then exp output.

#define NROWS_T 1024
#define NCOLS_T 1024
#define CHUNKS 16
#define CHUNK_ROWS (NROWS_T / CHUNKS)   // 64
#define BIGP (1.0e30f)                  // sentinel: x - BIGP -> exp underflows to 0

typedef __attribute__((ext_vector_type(2))) float v2f;
typedef __attribute__((ext_vector_type(8))) float v8f;

#if defined(__AMDGCN__) && __has_builtin(__builtin_amdgcn_wmma_f32_16x16x4_f32)
#define SINK_HAVE_WMMA 1
#else
#define SINK_HAVE_WMMA 0
#endif

#if defined(__AMDGCN__) && __has_builtin(__builtin_amdgcn_global_load_async_to_lds_b128) && \
    __has_builtin(__builtin_amdgcn_s_wait_asynccnt)
#define SINK_HAVE_ASYNC 1
#else
#define SINK_HAVE_ASYNC 0
#endif

#if SINK_HAVE_ASYNC
// Builtin signature (probe-confirmed): (addrspace(1) v4i*, addrspace(3) v4i*, Ii, Ii)
typedef int v4i __attribute__((vector_size(16)));
typedef __attribute__((address_space(1))) v4i glb_v4i;
typedef __attribute__((address_space(3))) v4i lds_v4i;
// 16B per lane, global -> LDS, tracked by ASYNCcnt.
__device__ __forceinline__ void sink_async_copy16(const float* gsrc, float* ldst) {
    __builtin_amdgcn_global_load_async_to_lds_b128(
        (glb_v4i*)(uintptr_t)gsrc, (lds_v4i*)(uintptr_t)ldst,
        /*imm offset=*/0, /*cpol=*/0);
}
#endif

// ---------------------------------------------------------------- init u,v=0
__global__ void sink_init_kernel(float* __restrict__ uv, int n) {
    int i = blockIdx.x * blockDim.x + threadIdx.x;
    if (i < n) uv[i] = 0.0f;
}

// ------------------------------------------------------- row pass: update u
// u[i] = u_old[i] + log( sum_{j<nc} exp(m[i][j] - v[j] - u_old[i]) )
// wave-per-row; lanes load float4 (512B per wave per trip, coalesced).
// Column masking is baked into the staged vs[] (sentinel +1e30) -> branchless.
__global__ void sink_row_pass(const float* __restrict__ m,
                              const int* __restrict__ nrows,
                              const int* __restrict__ ncols,
                              float* __restrict__ u,
                              const float* __restrict__ v) {
    const int b    = blockIdx.y;
    const int lane = threadIdx.x & 31;
    const int wave = threadIdx.x >> 5;
    const int r    = blockIdx.x * (blockDim.x >> 5) + wave;
    const int nr   = nrows[b];
    const int nc   = ncols[b];
    const int tid  = threadIdx.x;

    __shared__ __align__(16) float vs[NCOLS_T];
#if SINK_HAVE_ASYNC
    sink_async_copy16(v + (b << 10) + 4 * tid, &vs[4 * tid]);
    __builtin_amdgcn_s_wait_asynccnt(0);
    __syncthreads();
    for (int j = nc + tid; j < NCOLS_T; j += 256) vs[j] = BIGP;   // sentinel tail
#else
    for (int j = tid; j < NCOLS_T; j += 256)
        vs[j] = (j < nc) ? v[(b << 10) + j] : BIGP;
#endif
    __syncthreads();

    if (r >= nr) return;

    const float* row = m + (((size_t)b) << 20) + (((size_t)r) << 10);
    const float  s   = u[(b << 10) + r];   // shift by old potential: stable lse

    float acc = 0.0f;
    for (int j4 = lane << 2; j4 < nc; j4 += 128) {
        const float4 x  = *(const float4*)(row + j4);
        const float4 vv = *(const float4*)(&vs[j4]);
        // no selects: sentinel vs[] drives masked components to exp(-huge)=0
        acc += __expf(x.x - vv.x - s) + __expf(x.y - vv.y - s) +
               __expf(x.z - vv.z - s) + __expf(x.w - vv.w - s);
    }

#pragma unroll
    for (int mask = 16; mask > 0; mask >>= 1)
        acc += __shfl_xor(acc, mask, 32);

    if (lane == 0)
        u[(b << 10) + r] = s + __logf(acc);
}

// --------------------------------------- col pass A: partial column sums
// partial[b][chunk][j] = sum_{i in chunk, valid} exp(m[i][j] - u[i] - v_old[j])
// One wave owns 16 columns at a time; rows consumed 16 at a time through two
// independent V_WMMA_F32_16X16X4_F32 accumulators (A = ones(16x4), B = 4x16
// exp-tile):  D[m][n] = sum_k B[k][n]  -> column sums in every row of D.
// All validity masking is baked into staged us[]/vs[] sentinels -> the inner
// loop is pure load/sub/exp/wmma with no EXEC manipulation.
__global__ void sink_col_partial(const float* __restrict__ m,
                                 const int* __restrict__ nrows,
                                 const int* __restrict__ ncols,
                                 const float* __restrict__ u,
                                 const float* __restrict__ v,
                                 float* __restrict__ partial) {
    const int b     = blockIdx.y;
    const int chunk = blockIdx.x;             // 16 chunks of 64 rows
    const int rbase = chunk * CHUNK_ROWS;
    const int lane  = threadIdx.x & 31;
    const int wave  = threadIdx.x >> 5;       // 8 waves per block
    const int nr    = nrows[b];
    const int nc    = ncols[b];
    const int tid   = threadIdx.x;

    __shared__ __align__(16) float us[CHUNK_ROWS];
    __shared__ __align__(16) float vs[NCOLS_T];
#if SINK_HAVE_ASYNC
    sink_async_copy16(v + (b << 10) + 4 * tid, &vs[4 * tid]);
    if (tid < CHUNK_ROWS / 4)
        sink_async_copy16(u + (b << 10) + rbase + 4 * tid, &us[4 * tid]);
    __builtin_amdgcn_s_wait_asynccnt(0);
    __syncthreads();
    for (int j = nc + tid; j < NCOLS_T; j += 256) vs[j] = BIGP;   // sentinel tail
    {
        int kv = nr - rbase; if (kv < 0) kv = 0;
        for (int i = kv + tid; i < CHUNK_ROWS; i += 256) us[i] = BIGP;
    }
#else
    if (tid < CHUNK_ROWS)
        us[tid] = (rbase + tid < nr) ? u[(b << 10) + rbase + tid] : BIGP;
    for (int j = tid; j < NCOLS_T; j += 256)
        vs[j] = (j < nc) ? v[(b << 10) + j] : BIGP;
#endif
    __syncthreads();

    // uniform per block: valid rows in this chunk, rounded to 16
    int kvalid = nr - rbase;
    if (kvalid > CHUNK_ROWS) kvalid = CHUNK_ROWS;
    const int kend = (kvalid > 0) ? ((kvalid + 15) & ~15) : 0;

    // B-matrix element position for this lane (ISA 7.12.2 striping):
    //   VGPR0: lanes 0-15 -> K row 0, lanes 16-31 -> K row 2 (col = lane&15)
    //   VGPR1: K rows 1 / 3.
    const int rowoff = (lane < 16) ? 0 : 2;
    const float* mb = m + (((size_t)b) << 20);

    for (int cg = wave; cg < (NCOLS_T / 16); cg += 8) {
        const int   col = (cg << 4) + (lane & 15);
        const float vc  = vs[col];              // +1e30 sentinel if col invalid
        const float* pb = mb + (((size_t)(rbase + rowoff)) << 10) + col;

#if SINK_HAVE_WMMA
        v8f c0 = {}, c1 = {};
        const v2f ones = {1.0f, 1.0f};
#else
        float acc = 0.0f;
#endif
        for (int k = 0; k < kend; k += 16) {
#pragma unroll
            for (int t = 0; t < 4; ++t) {       // 4 tiles of 4 rows
                const int kk = k + 4 * t;
                const float x0 = pb[((size_t)kk << 10)];
                const float x1 = pb[((size_t)kk << 10) + NCOLS_T];
                const float a0 = x0 - us[kk + rowoff]     - vc;
                const float a1 = x1 - us[kk + rowoff + 1] - vc;
#if SINK_HAVE_WMMA
                v2f bt = {__expf(a0), __expf(a1)};
                if (t & 1)
                    c1 = __builtin_amdgcn_wmma_f32_16x16x4_f32(false, ones, false, bt,
                                                               (short)0, c1, false, false);
                else
                    c0 = __builtin_amdgcn_wmma_f32_16x16x4_f32(false, ones, false, bt,
                                                               (short)0, c0, false, false);
#else
                acc += __expf(a0) + __expf(a1);
#endif
            }
        }

#if SINK_HAVE_WMMA
        const float colsum = c0[0] + c1[0];  // D row M=0: VGPR0, per column n=lane&15
#else
        // lane L and L+16 hold the same column (different K rows): fold halves
        const float colsum = acc + __shfl_xor(acc, 16, 32);
#endif
        if (lane < 16)
            partial[(((size_t)(b * CHUNKS + chunk)) << 10) + col] = colsum;
    }
}

// --------------------------------------- col pass B: combine chunk partials
// v[j] += log( sum_chunks partial[b][chunk][j] )   for j < ncols
__global__ void sink_col_combine(const float* __restrict__ partial,
                                 const int* __restrict__ ncols,
                                 float* __restrict__ v) {
    const int b = blockIdx.y;
    const int j = blockIdx.x * blockDim.x + threadIdx.x;
    if (j >= ncols[b]) return;
    float s = 0.0f;
#pragma unroll
    for (int c = 0; c < CHUNKS; ++c)
        s += partial[(((size_t)(b * CHUNKS + c)) << 10) + j];
    v[(b << 10) + j] += __logf(s);
}

// ---------------------------------------------------------------- finalize
// out[b][i][j] = valid ? exp(m - u_i - v_j) : 0 ; float4 per thread,
// one block covers exactly one row (u_i uniform per block).
__global__ void sink_finalize(const float* __restrict__ m,
                              const int* __restrict__ nrows,
                              const int* __restrict__ ncols,
                              const float* __restrict__ u,
                              const float* __restrict__ v,
                              float* __restrict__ out) {
    const int b   = blockIdx.y;
    const int idx = blockIdx.x * blockDim.x + threadIdx.x;  // float4 index in batch
    const int i   = idx >> 8;              // 256 float4 per row
    const int j4  = (idx & 255) << 2;
    const int nr  = nrows[b];
    const int nc  = ncols[b];

    const size_t base = (((size_t)b) << 20) + (((size_t)i) << 10) + j4;
    const float4 x  = *(const float4*)(m + base);
    const float  ui = u[(b << 10) + i];
    const float4 vj = *(const float4*)(v + (b << 10) + j4);

    float4 o;
    const bool rok = i < nr;
    o.x = (rok && (j4 + 0 < nc)) ? __expf(x.x - ui - vj.x) : 0.0f;
    o.y = (rok && (j4 + 1 < nc)) ? __expf(x.y - ui - vj.y) : 0.0f;
    o.z = (rok && (j4 + 2 < nc)) ? __expf(x.z - ui - vj.z) : 0.0f;
    o.w = (rok && (j4 + 3 < nc)) ? __expf(x.w - ui - vj.w) : 0.0f;
    *(float4*)(out + base) = o;
}

// ---------------------------------------------------------------------------
extern "C" void kernel_launch(void* const* d_in, const int* in_sizes, int n_in,
                              void* d_out, int out_size, void* d_ws, size_t ws_size,
                              hipStream_t stream) {
    const float* m     = (const float*)d_in[0];
    const int*   nrows = (const int*)d_in[1];
    const int*   ncols = (const int*)d_in[2];
    float*       out   = (float*)d_out;

    const int B = in_sizes[1];   // 64

    // workspace layout
    float* u       = (float*)d_ws;                    // B*1024
    float* v       = u + (size_t)B * NROWS_T;         // B*1024
    float* partial = v + (size_t)B * NCOLS_T;         // B*CHUNKS*1024

    // u = v = 0
    {
        const int n = B * (NROWS_T + NCOLS_T);
        sink_init_kernel<<<(n + 255) / 256, 256, 0, stream>>>(u, n);
    }

    const dim3 rowGrid(NROWS_T / 8, B);      // 8 waves (rows) per 256-thread block
    const dim3 colGrid(CHUNKS, B);
    const dim3 cmbGrid(NCOLS_T / 256, B);

    for (int it = 0; it < 10; ++it) {
        sink_row_pass   <<<rowGrid, 256, 0, stream>>>(m, nrows, ncols, u, v);
        sink_col_partial<<<colGrid, 256, 0, stream>>>(m, nrows, ncols, u, v, partial);
        sink_col_combine<<<cmbGrid, 256, 0, stream>>>(partial, ncols, v);
    }

    const dim3 finGrid((NROWS_T * NCOLS_T) / 4 / 256, B);  // 1024 blocks x 256 thr
    sink_finalize<<<finGrid, 256, 0, stream>>>(m, nrows, ncols, u, v, out);
}